// SA_LUT3DGenerator_9612136808561
// MI455X (gfx1250) — compile-verified
//
#include <hip/hip_runtime.h>
#include <stdint.h>

// Problem constants (match reference)
#define NCOL   3
#define NV     33
#define NV2    (NV*NV)      // 1089
#define NV3    (NV*NV*NV)   // 35937
#define NR     8
#define BB     4
#define HH     512
#define WW     512
#define HWP    (HH*WW)                 // 262144 (2^18)
#define TOTAL_PIX (BB*HWP)             // 1048576

#define BLOCK  128
#define PPT    4                        // pixels per thread (float4 along W)
#define TILE_PIX (BLOCK*PPT)            // 512 pixels per workgroup

#define HAS_TDM __has_builtin(__builtin_amdgcn_tensor_load_to_lds)

typedef unsigned int u32x4 __attribute__((ext_vector_type(4)));
typedef int          i32x4 __attribute__((ext_vector_type(4)));
typedef int          i32x8 __attribute__((ext_vector_type(8)));

// compile-time-foldable float4 component select (loops are fully unrolled)
__device__ __forceinline__ float cmp4(const float4& v, int j) {
    return j == 0 ? v.x : (j == 1 ? v.y : (j == 2 ? v.z : v.w));
}

#if HAS_TDM
// Build D# and issue one TDM load of a [NR x TILE_PIX] f32 tile (row stride HWP
// elements) from `gaddr` into LDS byte offset `lds_off`.  Rows land consecutively
// in LDS: row r at lds_off + r*TILE_PIX*4 — identical to smX[r][tid] layout.
__device__ __forceinline__ void tdm_load_tile(uint64_t gaddr, uint32_t lds_off) {
    u32x4 g0;
    g0[0] = 1u;                                            // count=1, user mode
    g0[1] = lds_off;                                       // lds_addr (bytes)
    g0[2] = (uint32_t)gaddr;                               // global_addr[31:0]
    g0[3] = (uint32_t)((gaddr >> 32) & 0x01FFFFFFu)        // global_addr[56:32]
          | (2u << 30);                                    // type = 2 ("image")
    i32x8 g1;
    g1[0] = (2 << 16);                 // wg_mask=0, data_size=2 (4 bytes)
    g1[1] = 0;                         // atomic_barrier_addr=0, tensor_dim0[15:0]=0 (HWP low16)
    g1[2] = (int)((HWP >> 16) & 0xFFFF) | (NR << 16);  // tensor_dim0[31:16]=4, tensor_dim1[15:0]=8
    g1[3] = (TILE_PIX << 16);          // tensor_dim1[31:16]=0, tile_dim0=512
    g1[4] = NR;                        // tile_dim1=8, tile_dim2=0
    g1[5] = HWP;                       // tensor_dim0_stride[31:0]
    g1[6] = 0;                         // stride0[47:32]=0, stride1[15:0]=0
    g1[7] = 0;                         // stride1[47:16]=0
    i32x4 z4 = {0, 0, 0, 0};
#if __clang_major__ >= 23
    i32x8 z8 = {0, 0, 0, 0, 0, 0, 0, 0};
    __builtin_amdgcn_tensor_load_to_lds(g0, g1, z4, z4, z8, 0);
#else
    __builtin_amdgcn_tensor_load_to_lds(g0, g1, z4, z4, 0);
#endif
}
#endif

__global__ __launch_bounds__(BLOCK) void lut3d_fused_kernel(
    const float* __restrict__ ilut,   // (B, R, H, W)
    const float* __restrict__ plut,   // (B, R, H, W)
    const float* __restrict__ img,    // (B, 3, H, W)
    const float* __restrict__ lut,    // (3*33^3, 8) row-major: row = 8 contiguous floats
    float* __restrict__ out)          // (B, 3, H, W)
{
    __shared__ float4 smI[NR][BLOCK];   // staged ilut tile (16 KB), row r = 512 floats
    __shared__ float4 smP[NR][BLOCK];   // staged plut tile (16 KB)

    const int tid = threadIdx.x;
    const long long pix0 = (long long)blockIdx.x * TILE_PIX + (long long)tid * PPT;
    const int b  = (int)(pix0 >> 18);           // HWP = 2^18; constant per block
    const int p  = (int)(pix0 & (HWP - 1));
    const int p0 = (int)(((long long)blockIdx.x * TILE_PIX) & (HWP - 1)); // block tile start

    // ---- Warm L2 with the LUT table (global_prefetch_b8). 3.45 MB = 26953 x 128B lines.
    {
        const long long t = (long long)blockIdx.x * BLOCK + tid;
        const long long lut_lines = ((long long)NCOL * NV3 * NR * 4 + 127) / 128;
        if (t < lut_lines)
            __builtin_prefetch(lut + t * 32, 0, 1);   // 32 floats = one 128B line
    }

#if HAS_TDM
    // ---- Tensor Data Mover: one DMA per tensor stages the whole 32 KB tile,
    // keeping the VMEM pipe free for the LUT gathers. Wave 0 issues + waits
    // TENSORcnt, then the workgroup barrier publishes the LDS tile.
    if (tid < 32) {
        tdm_load_tile((uint64_t)(uintptr_t)(ilut + ((size_t)b * NR) * HWP + p0),
                      (uint32_t)(uintptr_t)&smI[0][0]);
        tdm_load_tile((uint64_t)(uintptr_t)(plut + ((size_t)b * NR) * HWP + p0),
                      (uint32_t)(uintptr_t)&smP[0][0]);
    }
#else
    // ---- Fallback: per-lane CDNA5 async global->LDS (ASYNCcnt), thread-local tiles.
    {
        const uint64_t ibase = (uint64_t)(uintptr_t)(ilut + ((size_t)b * NR) * HWP + p);
        const uint64_t pbase = (uint64_t)(uintptr_t)(plut + ((size_t)b * NR) * HWP + p);
        #pragma unroll
        for (int r = 0; r < NR; ++r) {
            uint32_t lofI = (uint32_t)(uintptr_t)&smI[r][tid];
            uint32_t lofP = (uint32_t)(uintptr_t)&smP[r][tid];
            uint64_t gI = ibase + (uint64_t)r * ((uint64_t)HWP * 4);
            uint64_t gP = pbase + (uint64_t)r * ((uint64_t)HWP * 4);
            asm volatile("global_load_async_to_lds_b128 %0, %1, off"
                         :: "v"(lofI), "v"(gI) : "memory");
            asm volatile("global_load_async_to_lds_b128 %0, %1, off"
                         :: "v"(lofP), "v"(gP) : "memory");
        }
    }
#endif

    // ---- Stream the image (overlaps with the DMA in flight).
    const float4 ix = *(const float4*)(img + ((size_t)b * NCOL + 0) * HWP + p);
    const float4 iy = *(const float4*)(img + ((size_t)b * NCOL + 1) * HWP + p);
    const float4 iz = *(const float4*)(img + ((size_t)b * NCOL + 2) * HWP + p);

    float xs[PPT], ys[PPT], zs[PPT];
    #pragma unroll
    for (int j = 0; j < PPT; ++j) { xs[j] = cmp4(ix, j); ys[j] = cmp4(iy, j); zs[j] = cmp4(iz, j); }

    // ---- Complete staging, then per-pixel rank weights s = ilut*plut.
#if HAS_TDM
    if (tid < 32) __builtin_amdgcn_s_wait_tensorcnt(0);
    __syncthreads();
#else
    asm volatile("s_wait_asynccnt 0x0" ::: "memory");
#endif

    float4 s4[NR];
    #pragma unroll
    for (int r = 0; r < NR; ++r) {
        float4 a = smI[r][tid];          // ds_load_b128
        float4 c = smP[r][tid];          // ds_load_b128
        s4[r] = make_float4(a.x * c.x, a.y * c.y, a.z * c.z, a.w * c.w);
    }

    float out_acc[NCOL][PPT];

    #pragma unroll
    for (int j = 0; j < PPT; ++j) {
        float xf = fminf(fmaxf(xs[j] * 32.0f, 0.0f), 32.0f);
        float yf = fminf(fmaxf(ys[j] * 32.0f, 0.0f), 32.0f);
        float zf = fminf(fmaxf(zs[j] * 32.0f, 0.0f), 32.0f);
        int x0 = min((int)xf, NV - 2);  float fx = xf - (float)x0;  float wx0 = 1.0f - fx;
        int y0 = min((int)yf, NV - 2);  float fy = yf - (float)y0;  float wy0 = 1.0f - fy;
        int z0 = min((int)zf, NV - 2);  float fz = zf - (float)z0;  float wz0 = 1.0f - fz;

        int zybase[4];
        zybase[0] = z0 * NV2 + y0 * NV + x0;
        zybase[1] = zybase[0] + NV;                 // y0+1
        zybase[2] = zybase[0] + NV2;                // z0+1
        zybase[3] = zybase[0] + NV2 + NV;           // z0+1, y0+1
        float wzy[4];
        wzy[0] = wz0 * wy0;  wzy[1] = wz0 * fy;  wzy[2] = fz * wy0;  wzy[3] = fz * fy;

        float s[NR];
        #pragma unroll
        for (int r = 0; r < NR; ++r) s[r] = cmp4(s4[r], j);

        #pragma unroll
        for (int c = 0; c < NCOL; ++c) {
            const float* lutc = lut + (size_t)c * (NV3 * NR);

            // Issue all 16 b128 gathers for this (pixel,color) before the FMA chains.
            float4 rg[4][4];
            #pragma unroll
            for (int k = 0; k < 4; ++k) {
                const float4* rp = reinterpret_cast<const float4*>(lutc + (size_t)zybase[k] * NR);
                rg[k][0] = rp[0];  rg[k][1] = rp[1];   // row x0 (8 floats)
                rg[k][2] = rp[2];  rg[k][3] = rp[3];   // row x0+1 (adjacent, contiguous)
            }

            float acc = 0.0f;
            #pragma unroll
            for (int k = 0; k < 4; ++k) {
                float4 r0 = rg[k][0], r1 = rg[k][1], r2 = rg[k][2], r3 = rg[k][3];
                float d0 = fmaf(r0.x, s[0], fmaf(r0.y, s[1], fmaf(r0.z, s[2], fmaf(r0.w, s[3],
                           fmaf(r1.x, s[4], fmaf(r1.y, s[5], fmaf(r1.z, s[6], r1.w * s[7])))))));
                float d1 = fmaf(r2.x, s[0], fmaf(r2.y, s[1], fmaf(r2.z, s[2], fmaf(r2.w, s[3],
                           fmaf(r3.x, s[4], fmaf(r3.y, s[5], fmaf(r3.z, s[6], r3.w * s[7])))))));
                acc = fmaf(wzy[k], fmaf(wx0, d0, fx * d1), acc);
            }
            out_acc[c][j] = acc;
        }
    }

    // ---- Store (B, 3, H, W), float4-coalesced.
    #pragma unroll
    for (int c = 0; c < NCOL; ++c) {
        float4 o = make_float4(out_acc[c][0], out_acc[c][1], out_acc[c][2], out_acc[c][3]);
        *reinterpret_cast<float4*>(out + ((size_t)b * NCOL + c) * HWP + p) = o;
    }
}

extern "C" void kernel_launch(void* const* d_in, const int* in_sizes, int n_in,
                              void* d_out, int out_size, void* d_ws, size_t ws_size,
                              hipStream_t stream) {
    const float* ilut = (const float*)d_in[0];
    const float* plut = (const float*)d_in[1];
    const float* img  = (const float*)d_in[2];
    const float* lut  = (const float*)d_in[3];
    float* out = (float*)d_out;

    dim3 grid(TOTAL_PIX / TILE_PIX);   // 2048 blocks
    dim3 block(BLOCK);                 // 128 threads = 4 waves (wave32)
    hipLaunchKernelGGL(lut3d_fused_kernel, grid, block, 0, stream,
                       ilut, plut, img, lut, out);
}